// T5DecoderWrapper_52312701665470
// MI455X (gfx1250) — compile-verified
//
#include <hip/hip_runtime.h>
#include <hip/hip_bf16.h>
#include <math.h>

// Problem constants (from reference): B=16, H=16, DH=64, L=2048, DM=1024
#define BATCH 16
#define NH    16
#define DHD   64
#define SEQ   2048
#define DMODEL 1024

typedef __attribute__((ext_vector_type(2))) float v2f;
typedef __attribute__((ext_vector_type(4))) float v4f;
typedef __attribute__((ext_vector_type(8))) float v8f;

// ---------------------------------------------------------------------------
// WMMA f32 GEMM: out(16 x 1024) = A(16 x 1024) @ W(1024 x 1024)
// One wave (32 threads) per 16x16 output tile; 64 tiles along N.
// Uses V_WMMA_F32_16X16X4_F32 (fp32 in / fp32 accumulate — matches reference).
// ---------------------------------------------------------------------------
__global__ __launch_bounds__(32) void wmma_gemm16(const float* __restrict__ A,
                                                  const float* __restrict__ W,
                                                  float* __restrict__ out) {
    const int n0   = blockIdx.x * 16;      // output column tile
    const int lane = threadIdx.x;          // 0..31, full wave32
    const int lo   = lane & 15;
    const int hi   = lane >> 4;            // 0 or 1 (half-wave)

    v8f acc = {};
    const float* arow = A + lo * DMODEL;   // A row m = lo (both halves index M by lo)
    const float* wcol = W + n0 + lo;       // W column n0+lo

    for (int k = 0; k < DMODEL; k += 4) {
        v2f a, b;
        // A 16x4 fragment: lane(hi,lo) holds A[lo][k + 2*hi + {0,1}]
        a.x = arow[k + 2 * hi];
        a.y = arow[k + 2 * hi + 1];
        // B 4x16 fragment: VGPR j holds row k + 2*hi + j, column n0+lo
        b.x = wcol[(long)(k + 2 * hi) * DMODEL];
        b.y = wcol[(long)(k + 2 * hi + 1) * DMODEL];
        acc = __builtin_amdgcn_wmma_f32_16x16x4_f32(
            /*neg_a=*/false, a, /*neg_b=*/false, b,
            /*c_mod=*/(short)0, acc, /*reuse_a=*/false, /*reuse_b=*/false);
    }

    // D 16x16: VGPR i -> row (i + 8*hi), col (n0 + lo)
#pragma unroll
    for (int i = 0; i < 8; ++i) {
        out[(long)(i + 8 * hi) * DMODEL + n0 + lo] = acc[i];
    }
}

// ---------------------------------------------------------------------------
// Bulk KV-cache copy: past_key/past_value -> key_states/value_states outputs.
// Loads use default RT policy (populates L2 for the attention kernel, which
// re-reads past K/V). Stores are NON-TEMPORAL (TH=NT) so 256 MB of written
// output does not evict the read data from the 192 MB L2.
// ---------------------------------------------------------------------------
__global__ __launch_bounds__(256) void kv_copy(const v4f* __restrict__ pk,
                                               const v4f* __restrict__ pv,
                                               v4f* __restrict__ ok,
                                               v4f* __restrict__ ov,
                                               long n4) {
    long i = (long)blockIdx.x * blockDim.x + threadIdx.x;
    const long stride = (long)gridDim.x * blockDim.x;
    for (; i < n4; i += stride) {
        v4f k = pk[i];               // RT load -> stays in L2 for attention
        v4f v = pv[i];
        __builtin_nontemporal_store(k, &ok[i]);   // NT store, bypass L2 residency
        __builtin_nontemporal_store(v, &ov[i]);
    }
}

// ---------------------------------------------------------------------------
// Scatter the freshly projected k_new/v_new rows into the output caches
// at row cache_position[b] (per batch, all heads).
// ---------------------------------------------------------------------------
__global__ __launch_bounds__(256) void kv_scatter(const float* __restrict__ kn,
                                                  const float* __restrict__ vn,
                                                  const int* __restrict__ cpos,
                                                  float* __restrict__ ok,
                                                  float* __restrict__ ov) {
    const int t = blockIdx.x * blockDim.x + threadIdx.x;  // 16*16*64 = 16384
    const int d = t & (DHD - 1);
    const int h = (t >> 6) & (NH - 1);
    const int b = t >> 10;
    const int pos = cpos[b];
    const long dst = (((long)(b * NH + h)) * SEQ + pos) * DHD + d;
    const int src = b * DMODEL + h * DHD + d;
    ok[dst] = kn[src];
    ov[dst] = vn[src];
}

// ---------------------------------------------------------------------------
// Single-query attention per (b,h): scores = q.K^T + bias, softmax, ctx = p.V
// Reads past_key/past_value directly and substitutes the updated row l==pos,
// so it does not depend on the copy kernel and can reuse its L2 residency.
// ---------------------------------------------------------------------------
__global__ __launch_bounds__(256) void attn_kernel(const float* __restrict__ q,
                                                   const float* __restrict__ kn,
                                                   const float* __restrict__ vn,
                                                   const float* __restrict__ pk,
                                                   const float* __restrict__ pv,
                                                   const float* __restrict__ bias,
                                                   const int* __restrict__ cpos,
                                                   float* __restrict__ ctx) {
    __shared__ float q_sh[DHD];
    __shared__ float sc[SEQ];          // 8 KB
    __shared__ float red[256];
    __shared__ float cpart[4][DHD];

    const int bh = blockIdx.x;         // 0..255
    const int b = bh >> 4;
    const int h = bh & 15;
    const int t = threadIdx.x;
    const int pos = cpos[b];

    if (t < DHD) q_sh[t] = q[b * DMODEL + h * DHD + t];
    __syncthreads();

    const float* kbase = pk + (long)bh * SEQ * DHD;
    const float* knrow = kn + b * DMODEL + h * DHD;

    // Stage 1: scores + running max
    float lmax = -INFINITY;
    for (int l = t; l < SEQ; l += 256) {
        const float* krow = (l == pos) ? knrow : (kbase + (long)l * DHD);
        float dot = 0.f;
#pragma unroll
        for (int d = 0; d < DHD; d += 4) {
            float4 k4 = *(const float4*)(krow + d);
            dot += q_sh[d] * k4.x + q_sh[d + 1] * k4.y +
                   q_sh[d + 2] * k4.z + q_sh[d + 3] * k4.w;
        }
        const float s = dot + bias[(long)bh * SEQ + l];
        sc[l] = s;
        lmax = fmaxf(lmax, s);
    }

    // Stage 2a: max reduction
    red[t] = lmax;
    __syncthreads();
    for (int off = 128; off > 0; off >>= 1) {
        if (t < off) red[t] = fmaxf(red[t], red[t + off]);
        __syncthreads();
    }
    const float m = red[0];
    __syncthreads();

    // Stage 2b: exp + sum reduction
    float lsum = 0.f;
    for (int l = t; l < SEQ; l += 256) {
        const float e = expf(sc[l] - m);
        sc[l] = e;
        lsum += e;
    }
    red[t] = lsum;
    __syncthreads();
    for (int off = 128; off > 0; off >>= 1) {
        if (t < off) red[t] += red[t + off];
        __syncthreads();
    }
    const float inv = 1.0f / red[0];

    // Stage 3: ctx[d] = inv * sum_l p[l] * V[l][d]
    // 4 groups of 64 lanes; for fixed l, lanes d=0..63 read consecutive
    // addresses of V -> fully coalesced streaming.
    const int c = t >> 6;
    const int d = t & (DHD - 1);
    const float* vbase = pv + (long)bh * SEQ * DHD;
    const float* vnrow = vn + b * DMODEL + h * DHD;
    float acc = 0.f;
    for (int l = c * (SEQ / 4); l < (c + 1) * (SEQ / 4); ++l) {
        const float* vrow = (l == pos) ? vnrow : (vbase + (long)l * DHD);
        acc += sc[l] * vrow[d];
    }
    cpart[c][d] = acc;
    __syncthreads();
    if (c == 0) {
        const float tot = (cpart[0][d] + cpart[1][d] + cpart[2][d] + cpart[3][d]) * inv;
        ctx[b * DMODEL + h * DHD + d] = tot;
    }
}

// ---------------------------------------------------------------------------
// Launch: QKV projections (WMMA) -> cache copy -> scatter -> attention ->
//         output projection (WMMA). All on `stream`, graph-capture safe.
// ---------------------------------------------------------------------------
extern "C" void kernel_launch(void* const* d_in, const int* in_sizes, int n_in,
                              void* d_out, int out_size, void* d_ws, size_t ws_size,
                              hipStream_t stream) {
    (void)in_sizes; (void)n_in; (void)out_size; (void)ws_size;

    const float* hidden = (const float*)d_in[0];  // (16,1,1024)
    const float* pbias  = (const float*)d_in[1];  // (16,16,1,2048)
    const float* pkey   = (const float*)d_in[2];  // (16,16,2048,64)
    const float* pval   = (const float*)d_in[3];  // (16,16,2048,64)
    const int*   cpos   = (const int*)d_in[4];    // (16,1)
    const float* Wq     = (const float*)d_in[5];
    const float* Wk     = (const float*)d_in[6];
    const float* Wv     = (const float*)d_in[7];
    const float* Wo     = (const float*)d_in[8];

    float* out_attn = (float*)d_out;                               // 16*1024
    float* out_key  = out_attn + BATCH * DMODEL;                   // 16*16*2048*64
    float* out_val  = out_key + (long)BATCH * NH * SEQ * DHD;

    float* ws     = (float*)d_ws;                 // 4 * 16K floats = 256 KB
    float* q_ws   = ws;
    float* kn_ws  = ws + 1 * BATCH * DMODEL;
    float* vn_ws  = ws + 2 * BATCH * DMODEL;
    float* ctx_ws = ws + 3 * BATCH * DMODEL;

    // QKV projections (WMMA fp32)
    wmma_gemm16<<<DMODEL / 16, 32, 0, stream>>>(hidden, Wq, q_ws);
    wmma_gemm16<<<DMODEL / 16, 32, 0, stream>>>(hidden, Wk, kn_ws);
    wmma_gemm16<<<DMODEL / 16, 32, 0, stream>>>(hidden, Wv, vn_ws);

    // KV cache copy (bandwidth-dominant; NT stores) + single-row scatter
    const long n4 = (long)BATCH * NH * SEQ * DHD / 4;  // 8,388,608 float4
    kv_copy<<<8192, 256, 0, stream>>>((const v4f*)pkey, (const v4f*)pval,
                                      (v4f*)out_key, (v4f*)out_val, n4);
    kv_scatter<<<(BATCH * NH * DHD) / 256, 256, 0, stream>>>(kn_ws, vn_ws, cpos,
                                                             out_key, out_val);

    // Attention (reads past K/V + new rows; independent of copy ordering)
    attn_kernel<<<BATCH * NH, 256, 0, stream>>>(q_ws, kn_ws, vn_ws, pkey, pval,
                                                pbias, cpos, ctx_ws);

    // Output projection (WMMA fp32)
    wmma_gemm16<<<DMODEL / 16, 32, 0, stream>>>(ctx_ws, Wo, out_attn);
}